// s6ClassicModule_24154896073581
// MI455X (gfx1250) — compile-verified
//
#include <hip/hip_runtime.h>
#include <math.h>

// ---- problem constants (fixed by the reference) ----
#define NB 2      // batch
#define DD 1024   // d_model
#define LL 2048   // sequence length
#define NS 16     // D_STATE
#define RR 64     // DT_RANK
#define EE 96     // r + 2n rows of x_proj
#define NC 32     // scan chunks
#define LC 64     // scan chunk length (NC*LC == LL)

typedef float v2f __attribute__((ext_vector_type(2)));
typedef float v8f __attribute__((ext_vector_type(8)));

__device__ __forceinline__ float sigmoidf_(float x) { return 1.0f / (1.0f + expf(-x)); }
__device__ __forceinline__ float softplusf_(float x) { return (x > 20.0f) ? x : log1pf(expf(x)); }

// =====================================================================
// Kernel 1: x_dbl[b,e,l] = sum_d x_proj_w[e,d] * u[b,d,l]      (M=96, K=1024)
// =====================================================================
__global__ __launch_bounds__(128)
void k_xproj(const float* __restrict__ W, const float* __restrict__ U,
             float* __restrict__ Xd)
{
    __shared__ float As[16 * 68];
    const int tid  = threadIdx.x;
    const int lane = tid & 31;
    const int wv   = tid >> 5;
    const int m0   = blockIdx.y << 4;
    const int n0   = (blockIdx.x << 6) + (wv << 4);
    const int bi   = blockIdx.z;
    const int am   = lane & 15;          // M (A) / N (B,C) index within tile
    const int ak   = (lane >> 4) << 1;   // K pair base: 0 or 2
    v8f acc = {};
    const float* Ub = U + ((size_t)bi * DD) * LL + n0 + am;
    for (int kc = 0; kc < DD; kc += 64) {
        for (int i = tid; i < 16 * 64; i += 128) {
            const int r = i >> 6, c = i & 63;
            As[r * 68 + c] = W[(size_t)(m0 + r) * DD + kc + c];
        }
        __syncthreads();
        #pragma unroll
        for (int k = 0; k < 64; k += 4) {
            v2f a, b;
            a.x = As[am * 68 + k + ak];
            a.y = As[am * 68 + k + ak + 1];
            const float* bp = Ub + (size_t)(kc + k + ak) * LL;
            b.x = bp[0];
            b.y = bp[LL];
            acc = __builtin_amdgcn_wmma_f32_16x16x4_f32(
                false, a, false, b, (short)0, acc, false, false);
        }
        __syncthreads();
    }
    const int cn = n0 + am;
    const int cr = (lane >> 4) << 3;
    #pragma unroll
    for (int v = 0; v < 8; ++v)
        Xd[((size_t)bi * EE + m0 + cr + v) * LL + cn] = acc[v];
}

// =====================================================================
// Kernel 2: delta[b,d,l] = softplus( dt_proj_w @ x_dbl[:r] + bias )
// =====================================================================
__global__ __launch_bounds__(128)
void k_dt(const float* __restrict__ Wdt, const float* __restrict__ Xd,
          const float* __restrict__ bias, float* __restrict__ Delta)
{
    __shared__ float As[16 * 68];
    const int tid  = threadIdx.x;
    const int lane = tid & 31;
    const int wv   = tid >> 5;
    const int m0   = blockIdx.y << 4;
    const int n0   = (blockIdx.x << 6) + (wv << 4);
    const int bi   = blockIdx.z;
    const int am   = lane & 15;
    const int ak   = (lane >> 4) << 1;
    for (int i = tid; i < 16 * 64; i += 128) {
        const int r = i >> 6, c = i & 63;
        As[r * 68 + c] = Wdt[(size_t)(m0 + r) * RR + c];
    }
    __syncthreads();
    v8f acc = {};
    const float* Bb = Xd + ((size_t)bi * EE) * LL + n0 + am;
    #pragma unroll
    for (int k = 0; k < RR; k += 4) {
        v2f a, b;
        a.x = As[am * 68 + k + ak];
        a.y = As[am * 68 + k + ak + 1];
        const float* bp = Bb + (size_t)(k + ak) * LL;
        b.x = bp[0];
        b.y = bp[LL];
        acc = __builtin_amdgcn_wmma_f32_16x16x4_f32(
            false, a, false, b, (short)0, acc, false, false);
    }
    const int cn = n0 + am;
    const int cr = (lane >> 4) << 3;
    #pragma unroll
    for (int v = 0; v < 8; ++v) {
        const int row = m0 + cr + v;
        Delta[((size_t)bi * DD + row) * LL + cn] = softplusf_(acc[v] + bias[row]);
    }
}

// =====================================================================
// Chunked selective scan. Channel ch = (b,d); 16-lane half-wave = 16 states.
// Pass A: per (ch, chunk): local scan from h=0 -> summary S, and sum(delta).
//         (prod of dA over chunk == exp(A_n * sum delta) -- exact)
// Pass B: per ch: 32-step combine (exps hoisted, chain is pure FMA) -> Hin
// Pass C: per (ch, chunk): replay from Hin, emit y with fused tail.
// Critical path: 64 + 32 + 64 steps instead of 2048.
// =====================================================================
__global__ __launch_bounds__(256)
void k_scanA(const float* __restrict__ U, const float* __restrict__ Xd,
             const float* __restrict__ Delta, const float* __restrict__ Alog,
             float* __restrict__ S, float* __restrict__ SumD)
{
    const int lane = threadIdx.x & 31;
    const int gw   = blockIdx.x * (blockDim.x >> 5) + (threadIdx.x >> 5);
    const int p    = (gw << 1) + (lane >> 4);     // (ch, chunk) pair id
    const int ch   = p >> 5;                      // / NC
    const int c    = p & (NC - 1);
    const int bi   = ch >> 10;
    const int d    = ch & (DD - 1);
    const int n    = lane & 15;
    const int hb   = lane & 16;
    const float An = -expf(Alog[d * NS + n]);
    const size_t rowOff = ((size_t)bi * DD + d) * LL + (size_t)c * LC;
    const float* dRow = Delta + rowOff;
    const float* uRow = U + rowOff;
    const float* bRow = Xd + ((size_t)bi * EE + RR + n) * LL + (size_t)c * LC;
    float h = 0.0f, sdp = 0.0f;
    for (int l0 = 0; l0 < LC; l0 += 16) {
        const float dl = dRow[l0 + n];
        const float ul = uRow[l0 + n];
        sdp += dl;
        float bv[16];
        #pragma unroll
        for (int q = 0; q < 4; ++q) {
            float4 t = *(const float4*)(bRow + l0 + 4 * q);
            bv[4*q+0] = t.x; bv[4*q+1] = t.y; bv[4*q+2] = t.z; bv[4*q+3] = t.w;
        }
        #pragma unroll
        for (int j = 0; j < 16; ++j) {
            const float dj = __shfl(dl, hb + j, 32);
            const float uj = __shfl(ul, hb + j, 32);
            h = expf(dj * An) * h + dj * uj * bv[j];
        }
    }
    sdp += __shfl_xor(sdp, 8, 32);
    sdp += __shfl_xor(sdp, 4, 32);
    sdp += __shfl_xor(sdp, 2, 32);
    sdp += __shfl_xor(sdp, 1, 32);
    S[((size_t)ch * NC + c) * NS + n] = h;
    if (n == 0) SumD[(size_t)ch * NC + c] = sdp;
}

__global__ __launch_bounds__(256)
void k_scanB(const float* __restrict__ Alog, const float* __restrict__ S,
             const float* __restrict__ SumD, float* __restrict__ Hin)
{
    const int lane = threadIdx.x & 31;
    const int gw   = blockIdx.x * (blockDim.x >> 5) + (threadIdx.x >> 5);
    const int ch   = (gw << 1) + (lane >> 4);     // channel in [0, NB*DD)
    const int d    = ch & (DD - 1);
    const int n    = lane & 15;
    const float An = -expf(Alog[d * NS + n]);
    float P[NC];
    #pragma unroll
    for (int c = 0; c < NC; ++c)                  // independent -> parallel exps
        P[c] = expf(An * SumD[(size_t)ch * NC + c]);
    float hin = 0.0f;
    #pragma unroll
    for (int c = 0; c < NC; ++c) {                // serial chain: pure FMA
        Hin[((size_t)ch * NC + c) * NS + n] = hin;
        hin = P[c] * hin + S[((size_t)ch * NC + c) * NS + n];
    }
}

__global__ __launch_bounds__(256)
void k_scanC(const float* __restrict__ U, const float* __restrict__ Xd,
             const float* __restrict__ Delta, const float* __restrict__ Alog,
             const float* __restrict__ Dp, const float* __restrict__ Hin,
             float* __restrict__ Y)
{
    const int lane = threadIdx.x & 31;
    const int gw   = blockIdx.x * (blockDim.x >> 5) + (threadIdx.x >> 5);
    const int p    = (gw << 1) + (lane >> 4);
    const int ch   = p >> 5;
    const int c    = p & (NC - 1);
    const int bi   = ch >> 10;
    const int d    = ch & (DD - 1);
    const int n    = lane & 15;
    const int hb   = lane & 16;
    const float An = -expf(Alog[d * NS + n]);
    const float Dv = Dp[d];
    const size_t rowOff = ((size_t)bi * DD + d) * LL + (size_t)c * LC;
    const float* dRow = Delta + rowOff;
    const float* uRow = U + rowOff;
    const float* bRow = Xd + ((size_t)bi * EE + RR) * LL      + (size_t)n * LL + (size_t)c * LC;
    const float* cRow = Xd + ((size_t)bi * EE + RR + NS) * LL + (size_t)n * LL + (size_t)c * LC;
    float* yRow = Y + rowOff;
    float h = Hin[((size_t)ch * NC + c) * NS + n];
    for (int l0 = 0; l0 < LC; l0 += 16) {
        const float dl = dRow[l0 + n];
        const float ul = uRow[l0 + n];
        float bv[16], cv[16];
        #pragma unroll
        for (int q = 0; q < 4; ++q) {
            float4 t = *(const float4*)(bRow + l0 + 4 * q);
            bv[4*q+0] = t.x; bv[4*q+1] = t.y; bv[4*q+2] = t.z; bv[4*q+3] = t.w;
            float4 s = *(const float4*)(cRow + l0 + 4 * q);
            cv[4*q+0] = s.x; cv[4*q+1] = s.y; cv[4*q+2] = s.z; cv[4*q+3] = s.w;
        }
        float ymy = 0.0f;
        #pragma unroll
        for (int j = 0; j < 16; ++j) {
            const float dj = __shfl(dl, hb + j, 32);
            const float uj = __shfl(ul, hb + j, 32);
            h = expf(dj * An) * h + dj * uj * bv[j];
            float yv = h * cv[j];
            yv += __shfl_xor(yv, 8, 32);
            yv += __shfl_xor(yv, 4, 32);
            yv += __shfl_xor(yv, 2, 32);
            yv += __shfl_xor(yv, 1, 32);
            if (n == j) ymy = yv;                 // lane j keeps step j's result
        }
        float yo = ymy + ul * Dv;                 // + u*D
        yo = yo * (ul * sigmoidf_(ul));           // * silu(z), z == u
        yo = 0.5f * yo * (1.0f + erff(yo * 0.70710678118f)); // exact GELU
        yRow[l0 + n] = yo;
    }
}

// =====================================================================
// Kernel 4: out = out_w @ y + out_b ; result = a * sigmoid(g)
// Each wave: 16x32 output (two N tiles) x paired a/g rows ->
// 4 WMMAs per k-step sharing 2 A-frags + 2 B-frags.
// =====================================================================
__global__ __launch_bounds__(128)
void k_out(const float* __restrict__ Wo, const float* __restrict__ Yv,
           const float* __restrict__ bo, float* __restrict__ Out)
{
    __shared__ float Aa[16 * 68];
    __shared__ float Ag[16 * 68];
    const int tid  = threadIdx.x;
    const int lane = tid & 31;
    const int wv   = tid >> 5;
    const int m0   = blockIdx.y << 4;
    const int n0   = (blockIdx.x << 7) + (wv << 5);   // 128 cols/block, 32/wave
    const int bi   = blockIdx.z;
    const int am   = lane & 15;
    const int ak   = (lane >> 4) << 1;
    v8f aA0 = {}, aA1 = {}, aG0 = {}, aG1 = {};
    const float* Yb = Yv + ((size_t)bi * DD) * LL + n0 + am;
    for (int kc = 0; kc < DD; kc += 64) {
        for (int i = tid; i < 16 * 64; i += 128) {
            const int r = i >> 6, c = i & 63;
            Aa[r * 68 + c] = Wo[(size_t)(m0 + r) * DD + kc + c];
            Ag[r * 68 + c] = Wo[(size_t)(DD + m0 + r) * DD + kc + c];
        }
        __syncthreads();
        if (kc + 64 < DD)   // hint next weight chunk toward the caches
            __builtin_prefetch(&Wo[(size_t)(m0 + (tid >> 3)) * DD + kc + 64 + ((tid & 7) << 3)], 0, 1);
        #pragma unroll
        for (int k = 0; k < 64; k += 4) {
            v2f b0, b1, aa, ag;
            const float* bp = Yb + (size_t)(kc + k + ak) * LL;
            b0.x = bp[0];
            b0.y = bp[LL];
            b1.x = bp[16];
            b1.y = bp[LL + 16];
            aa.x = Aa[am * 68 + k + ak];
            aa.y = Aa[am * 68 + k + ak + 1];
            ag.x = Ag[am * 68 + k + ak];
            ag.y = Ag[am * 68 + k + ak + 1];
            aA0 = __builtin_amdgcn_wmma_f32_16x16x4_f32(false, aa, false, b0, (short)0, aA0, false, false);
            aA1 = __builtin_amdgcn_wmma_f32_16x16x4_f32(false, aa, false, b1, (short)0, aA1, false, false);
            aG0 = __builtin_amdgcn_wmma_f32_16x16x4_f32(false, ag, false, b0, (short)0, aG0, false, false);
            aG1 = __builtin_amdgcn_wmma_f32_16x16x4_f32(false, ag, false, b1, (short)0, aG1, false, false);
        }
        __syncthreads();
    }
    const int cn = n0 + am;
    const int cr = (lane >> 4) << 3;
    #pragma unroll
    for (int v = 0; v < 8; ++v) {
        const int row = m0 + cr + v;
        const float ba = bo[row];
        const float bg = bo[DD + row];
        float* o = Out + ((size_t)bi * DD + row) * LL + cn;
        o[0]  = (aA0[v] + ba) * sigmoidf_(aG0[v] + bg);
        o[16] = (aA1[v] + ba) * sigmoidf_(aG1[v] + bg);
    }
}

// =====================================================================
extern "C" void kernel_launch(void* const* d_in, const int* in_sizes, int n_in,
                              void* d_out, int out_size, void* d_ws, size_t ws_size,
                              hipStream_t stream)
{
    const float* u    = (const float*)d_in[0];  // (b,d,l)
    const float* xw   = (const float*)d_in[1];  // (96,d)
    const float* dtw  = (const float*)d_in[2];  // (d,64)
    const float* dtb  = (const float*)d_in[3];  // (d,)
    const float* alog = (const float*)d_in[4];  // (d,16)
    const float* Dp   = (const float*)d_in[5];  // (d,)
    const float* ow   = (const float*)d_in[6];  // (2d,d)
    const float* ob   = (const float*)d_in[7];  // (2d,)
    float* out = (float*)d_out;                 // (b,d,l)

    const size_t CH = (size_t)NB * DD;
    float* ws    = (float*)d_ws;
    float* xdbl  = ws;                              // NB*EE*LL
    float* delta = xdbl + (size_t)NB * EE * LL;     // NB*DD*LL
    float* yv    = delta + CH * LL;                 // NB*DD*LL
    float* Sst   = yv + CH * LL;                    // CH*NC*NS
    float* sumD  = Sst + CH * NC * NS;              // CH*NC
    float* hin   = sumD + CH * NC;                  // CH*NC*NS

    const dim3 blk(128);
    hipLaunchKernelGGL(k_xproj, dim3(LL / 64, EE / 16, NB), blk, 0, stream, xw, u, xdbl);
    hipLaunchKernelGGL(k_dt,    dim3(LL / 64, DD / 16, NB), blk, 0, stream, dtw, xdbl, dtb, delta);
    hipLaunchKernelGGL(k_scanA, dim3((CH * NC) / 16), dim3(256), 0, stream, u, xdbl, delta, alog, Sst, sumD);
    hipLaunchKernelGGL(k_scanB, dim3(CH / 16), dim3(256), 0, stream, alog, Sst, sumD, hin);
    hipLaunchKernelGGL(k_scanC, dim3((CH * NC) / 16), dim3(256), 0, stream, u, xdbl, delta, alog, Dp, hin, yv);
    hipLaunchKernelGGL(k_out,   dim3(LL / 128, DD / 16, NB), blk, 0, stream, ow, yv, ob, out);
}